// GCNBase_12335146074466
// MI455X (gfx1250) — compile-verified
//
#include <hip/hip_runtime.h>
#include <hip/hip_bf16.h>
#include <math.h>

typedef __attribute__((ext_vector_type(2))) float v2f;
typedef __attribute__((ext_vector_type(8))) float v8f;
typedef __attribute__((ext_vector_type(4))) unsigned int v4u;
typedef __attribute__((ext_vector_type(8))) int v8i;
typedef __attribute__((ext_vector_type(4))) int v4i;

#define EPS 1e-5f

__device__ __forceinline__ float elu1(float x) {
    return x > 0.0f ? x : (__expf(x) - 1.0f);
}

__device__ __forceinline__ void atomic_add_f32(float* p, float v) {
    __hip_atomic_fetch_add(p, v, __ATOMIC_RELAXED, __HIP_MEMORY_SCOPE_AGENT);
}

// ---------------------------------------------------------------- utilities
__global__ void zero_kernel(float* __restrict__ p, long n) {
    long i = blockIdx.x * (long)blockDim.x + threadIdx.x;
    if (i < n) p[i] = 0.0f;
}

// deg[col[e]] += 1
__global__ void deg_kernel(const int* __restrict__ ei, float* __restrict__ deg,
                           int E) {
    long e = blockIdx.x * (long)blockDim.x + threadIdx.x;
    if (e < E) atomic_add_f32(&deg[ei[E + e]], 1.0f);
}

// deg -> deg^(-1/2) (0 if deg==0)
__global__ void dis_kernel(float* __restrict__ dis, int N) {
    long i = blockIdx.x * (long)blockDim.x + threadIdx.x;
    if (i < N) {
        float d = dis[i];
        dis[i] = d > 0.0f ? __frsqrt_rn(fmaxf(d, 1.0f)) : 0.0f;
    }
}

// coef[e] = dis[row]*dis[col]*mask[e]
__global__ void coef_kernel(const int* __restrict__ ei, const float* __restrict__ dis,
                            const float* __restrict__ mask, float* __restrict__ coef,
                            int E) {
    long e = blockIdx.x * (long)blockDim.x + threadIdx.x;
    if (e < E) coef[e] = dis[ei[e]] * dis[ei[E + e]] * mask[e];
}

// X[n, 0:64]=x1, [64:128]=x2, [128:144]=rd   (packed ld=144)
__global__ void concat144_kernel(const float* __restrict__ x1, const float* __restrict__ x2,
                                 const float* __restrict__ rd, float* __restrict__ X,
                                 long total) {
    long i = blockIdx.x * (long)blockDim.x + threadIdx.x;
    if (i >= total) return;
    long n = i / 144;
    int c = (int)(i - n * 144);
    float v;
    if (c < 64)       v = x1[n * 64 + c];
    else if (c < 128) v = x2[n * 64 + (c - 64)];
    else              v = rd[n * 16 + (c - 128)];
    X[i] = v;
}

// X[n, 0:64]=x1, [64:128]=x2, [128:192]=elu(agg + bias)   (packed ld=192)
__global__ void conv_epilogue_kernel(const float* __restrict__ x1, const float* __restrict__ x2,
                                     const float* __restrict__ agg, const float* __restrict__ bias,
                                     float* __restrict__ X, long total) {
    long i = blockIdx.x * (long)blockDim.x + threadIdx.x;
    if (i >= total) return;
    long n = i / 192;
    int c = (int)(i - n * 192);
    float v;
    if (c < 64)       v = x1[n * 64 + c];
    else if (c < 128) v = x2[n * 64 + (c - 64)];
    else {
        int d = c - 128;
        v = elu1(agg[n * 64 + d] + bias[d]);
    }
    X[i] = v;
}

// ---------------------------------------------------------------- WMMA GEMM
// D[N, C] = A[N, K] * B[K, C]  (+bias, +elu) using V_WMMA_F32_16X16X4_F32.
// B (<=192x64 fp32, <=48KB) is staged once per block into LDS by the Tensor
// Data Mover; each block then sweeps mIter row strips against the LDS tile.
__global__ __launch_bounds__(256)
void wmma_gemm_kernel(const float* __restrict__ A, int lda,
                      const float* __restrict__ B, int ldb,
                      float* __restrict__ D, int ldd,
                      const float* __restrict__ bias, int do_elu,
                      int N, int K, int nTilesN, int mIter) {
    __shared__ float smemB[192 * 64];

    // ---- stage B[K, C] -> LDS via TDM (one wave issues; all waves sync) ----
    if (threadIdx.x < 32) {
        const unsigned long long gaddr = (unsigned long long)(const void*)B;
        const unsigned int lds_base = (unsigned int)(size_t)&smemB[0];
        const unsigned int C = (unsigned int)ldb;
        const unsigned int Ku = (unsigned int)K;

        v4u g0;
        g0.x = 1u;                                        // count=1, user D#
        g0.y = lds_base;                                  // lds_addr (bytes)
        g0.z = (unsigned int)(gaddr & 0xFFFFFFFFu);       // global_addr[31:0]
        g0.w = (unsigned int)((gaddr >> 32) & 0x1FFFFFFu) // global_addr[56:32]
             | (2u << 30);                                // type=2 (image)

        v8i g1;
        g1[0] = (int)(2u << 16);                          // data_size=2 -> 4B
        g1[1] = (int)((C & 0xFFFFu) << 16);               // tensor_dim0[15:0]
        g1[2] = (int)((C >> 16) | ((Ku & 0xFFFFu) << 16));// dim0[31:16], dim1[15:0]
        g1[3] = (int)((Ku >> 16) | (C << 16));            // dim1[31:16], tile_dim0=C
        g1[4] = (int)(Ku & 0xFFFFu);                      // tile_dim1=K, tile_dim2=0
        g1[5] = (int)C;                                   // tensor_dim0_stride[31:0]=C
        g1[6] = 0;                                        // stride hi / dim1_stride
        g1[7] = 0;

        v4i g2 = {0, 0, 0, 0};                            // unused (2-D tensor)
        v4i g3 = {0, 0, 0, 0};
        v8i g4 = {0, 0, 0, 0, 0, 0, 0, 0};                // 6-arg form extra group

        __builtin_amdgcn_tensor_load_to_lds(g0, g1, g2, g3, g4, 0);
        __builtin_amdgcn_s_wait_tensorcnt(0);             // data in LDS for this wave
    }
    __syncthreads();                                      // publish to all waves

    const int wid  = threadIdx.x >> 5;
    const int lane = threadIdx.x & 31;
    const int half = lane >> 4;     // 0: lanes 0-15, 1: lanes 16-31
    const int l16  = lane & 15;

    const int wavesM       = 8 / nTilesN;
    const int rowsPerBlock = wavesM * 16;
    const int tm   = wid / nTilesN;
    const int tn   = wid % nTilesN;
    const int col0 = tn * 16;
    const int cc   = col0 + l16;
    const float bv = bias ? bias[cc] : 0.0f;

    const float* bpl = &smemB[(size_t)(2 * half) * ldb + cc];

    for (int mt = 0; mt < mIter; ++mt) {
        const int row0 = (blockIdx.x * mIter + mt) * rowsPerBlock + tm * 16;

        int arow = row0 + l16;                 // clamp: EXEC stays all-ones
        if (arow >= N) arow = N - 1;
        const float* ap = A + (size_t)arow * lda + 2 * half;

        v8f c = {};
        for (int kk = 0; kk < K; kk += 4) {
            __builtin_prefetch(ap + kk + 32, 0, 3);       // global_prefetch_b8
            float2 av = *(const float2*)(ap + kk);
            v2f a, b;
            a.x = av.x;
            a.y = av.y;
            b.x = bpl[(size_t)kk * ldb];                  // ds_load from LDS
            b.y = bpl[(size_t)kk * ldb + ldb];
            c = __builtin_amdgcn_wmma_f32_16x16x4_f32(false, a, false, b,
                                                      (short)0, c, false, false);
        }

#pragma unroll
        for (int v = 0; v < 8; ++v) {
            int r = row0 + v + 8 * half;  // C/D layout: VGPR v -> M=v (+8 hi lanes)
            if (r < N) {
                float val = c[v] + bv;
                if (do_elu) val = elu1(val);
                D[(size_t)r * ldd + cc] = val;
            }
        }
    }
}

// ---------------------------------------------------------------- edge scatter
// AGG[col[e], :64] += coef[e] * H[row[e], :64]; one thread per (edge, 4 dims)
__global__ void scatter_kernel(const int* __restrict__ ei, const float* __restrict__ coef,
                               const float* __restrict__ H, float* __restrict__ AGG,
                               int E) {
    long idx = blockIdx.x * (long)blockDim.x + threadIdx.x;
    long total = (long)E * 16;
    if (idx >= total) return;
    int e  = (int)(idx >> 4);
    int d4 = (int)(idx & 15) * 4;
    int r  = ei[e];
    int cl = ei[E + e];
    float cf = coef[e];
    const float4 h = *(const float4*)(H + (size_t)r * 64 + d4);
    float* dst = AGG + (size_t)cl * 64 + d4;
    atomic_add_f32(dst + 0, cf * h.x);
    atomic_add_f32(dst + 1, cf * h.y);
    atomic_add_f32(dst + 2, cf * h.z);
    atomic_add_f32(dst + 3, cf * h.w);
}

// ---------------------------------------------------------------- batch norm
// blockDim.x == C; block processes 256 rows; column sums via atomics
__global__ void bn_stats_kernel(const float* __restrict__ X, int C, int N,
                                float* __restrict__ sums, float* __restrict__ sumsq) {
    const int c = threadIdx.x;
    const int r0 = blockIdx.x * 256;
    int r1 = r0 + 256;
    if (r1 > N) r1 = N;
    float s = 0.0f, s2 = 0.0f;
    for (int r = r0; r < r1; ++r) {
        float v = X[(size_t)r * C + c];
        s += v;
        s2 += v * v;
    }
    atomic_add_f32(&sums[c], s);
    atomic_add_f32(&sumsq[c], s2);
}

__global__ void bn_finalize_kernel(const float* __restrict__ sums, const float* __restrict__ sumsq,
                                   const float* __restrict__ g, const float* __restrict__ b,
                                   float* __restrict__ scale, float* __restrict__ shift,
                                   int C, int N) {
    int c = blockIdx.x * blockDim.x + threadIdx.x;
    if (c >= C) return;
    float inv_n = 1.0f / (float)N;
    float m = sums[c] * inv_n;
    float var = sumsq[c] * inv_n - m * m;
    float sc = g[c] * __frsqrt_rn(var + EPS);
    scale[c] = sc;
    shift[c] = b[c] - m * sc;
}

__global__ void bn_apply_kernel(float* __restrict__ X, const float* __restrict__ scale,
                                const float* __restrict__ shift, int C, long total) {
    long i = blockIdx.x * (long)blockDim.x + threadIdx.x;
    if (i >= total) return;
    int c = (int)(i % C);
    X[i] = X[i] * scale[c] + shift[c];
}

// ---------------------------------------------------------------- host driver
static inline int cdiv(long a, long b) { return (int)((a + b - 1) / b); }

extern "C" void kernel_launch(void* const* d_in, const int* in_sizes, int n_in,
                              void* d_out, int out_size, void* d_ws, size_t ws_size,
                              hipStream_t stream) {
    const float* x1   = (const float*)d_in[0];
    const float* x2   = (const float*)d_in[1];
    const float* rd   = (const float*)d_in[2];
    // d_in[3] = batch (unused by the computation)
    const float* mask = (const float*)d_in[4];
    const int*   ei   = (const int*)d_in[5];
    const float* conv0_w = (const float*)d_in[6];
    const float* conv0_b = (const float*)d_in[7];
    const float* conv1_w = (const float*)d_in[8];
    const float* conv1_b = (const float*)d_in[9];
    const float* bn0_g = (const float*)d_in[10];
    const float* bn0_b = (const float*)d_in[11];
    const float* bn1_g = (const float*)d_in[12];
    const float* bn1_b = (const float*)d_in[13];
    const float* mlp0_w = (const float*)d_in[14];
    const float* mlp0_b = (const float*)d_in[15];
    const float* mlp1_w = (const float*)d_in[16];
    const float* mlp1_b = (const float*)d_in[17];
    const float* mlp2_w = (const float*)d_in[18];
    const float* mlp2_b = (const float*)d_in[19];
    const float* bnm0_g = (const float*)d_in[20];
    const float* bnm0_b = (const float*)d_in[21];
    const float* bnm1_g = (const float*)d_in[22];
    const float* bnm1_b = (const float*)d_in[23];
    const float* bnm2_g = (const float*)d_in[24];
    const float* bnm2_b = (const float*)d_in[25];

    const int N = in_sizes[0] / 64;   // 80000
    const int E = in_sizes[4];        // 2560000

    // workspace carve-up (floats)
    float* ws = (float*)d_ws;
    size_t off = 0;
    float* coef = ws + off; off += (size_t)E;
    float* dis  = ws + off; off += (size_t)N;
    float* XA   = ws + off; off += (size_t)N * 192;
    float* XB   = ws + off; off += (size_t)N * 192;
    float* H    = ws + off; off += (size_t)N * 64;
    float* AGG  = ws + off; off += (size_t)N * 64;
    float* RED  = ws + off; off += 2 * 192;           // column sums / sumsq
    float* SC   = ws + off; off += 2 * 192;           // bn scale / shift
    float* sums = RED, *sumsq = RED + 192;
    float* scale = SC, *shift = SC + 192;

    float* OUT = (float*)d_out;       // [N, 32]

    const int T = 256;
    const int MI = 8;                 // row strips per block in the GEMM
    const long NE = (long)E;
    const long N144 = (long)N * 144, N192 = (long)N * 192, N64 = (long)N * 64;

    // --- normalization coefficients ------------------------------------
    zero_kernel<<<cdiv(N, T), T, 0, stream>>>(dis, N);
    deg_kernel<<<cdiv(NE, T), T, 0, stream>>>(ei, dis, E);
    dis_kernel<<<cdiv(N, T), T, 0, stream>>>(dis, N);
    coef_kernel<<<cdiv(NE, T), T, 0, stream>>>(ei, dis, mask, coef, E);

    // --- conv0: X[144] -> H[64] -> scatter -> XB[192] -> BN ------------
    concat144_kernel<<<cdiv(N144, T), T, 0, stream>>>(x1, x2, rd, XA, N144);
    wmma_gemm_kernel<<<cdiv(N, 32 * MI), T, 0, stream>>>(XA, 144, conv0_w, 64,
                                                         H, 64, nullptr, 0, N, 144, 4, MI);
    zero_kernel<<<cdiv(N64, T), T, 0, stream>>>(AGG, N64);
    scatter_kernel<<<cdiv(NE * 16, T), T, 0, stream>>>(ei, coef, H, AGG, E);
    conv_epilogue_kernel<<<cdiv(N192, T), T, 0, stream>>>(x1, x2, AGG, conv0_b, XB, N192);
    zero_kernel<<<2, 192, 0, stream>>>(RED, 2 * 192);
    bn_stats_kernel<<<cdiv(N, 256), 192, 0, stream>>>(XB, 192, N, sums, sumsq);
    bn_finalize_kernel<<<1, 192, 0, stream>>>(sums, sumsq, bn0_g, bn0_b, scale, shift, 192, N);
    bn_apply_kernel<<<cdiv(N192, T), T, 0, stream>>>(XB, scale, shift, 192, N192);

    // --- conv1: XB[192] -> H[64] -> scatter -> XA[192] -> BN -----------
    wmma_gemm_kernel<<<cdiv(N, 32 * MI), T, 0, stream>>>(XB, 192, conv1_w, 64,
                                                         H, 64, nullptr, 0, N, 192, 4, MI);
    zero_kernel<<<cdiv(N64, T), T, 0, stream>>>(AGG, N64);
    scatter_kernel<<<cdiv(NE * 16, T), T, 0, stream>>>(ei, coef, H, AGG, E);
    conv_epilogue_kernel<<<cdiv(N192, T), T, 0, stream>>>(x1, x2, AGG, conv1_b, XA, N192);
    zero_kernel<<<2, 192, 0, stream>>>(RED, 2 * 192);
    bn_stats_kernel<<<cdiv(N, 256), 192, 0, stream>>>(XA, 192, N, sums, sumsq);
    bn_finalize_kernel<<<1, 192, 0, stream>>>(sums, sumsq, bn1_g, bn1_b, scale, shift, 192, N);
    bn_apply_kernel<<<cdiv(N192, T), T, 0, stream>>>(XA, scale, shift, 192, N192);

    // --- mlp0: XA[192] -> H[64], elu+bias fused, then BN ---------------
    wmma_gemm_kernel<<<cdiv(N, 32 * MI), T, 0, stream>>>(XA, 192, mlp0_w, 64,
                                                         H, 64, mlp0_b, 1, N, 192, 4, MI);
    zero_kernel<<<2, 192, 0, stream>>>(RED, 2 * 192);
    bn_stats_kernel<<<cdiv(N, 256), 64, 0, stream>>>(H, 64, N, sums, sumsq);
    bn_finalize_kernel<<<1, 64, 0, stream>>>(sums, sumsq, bnm0_g, bnm0_b, scale, shift, 64, N);
    bn_apply_kernel<<<cdiv(N64, T), T, 0, stream>>>(H, scale, shift, 64, N64);

    // --- mlp1: H[64] -> AGG[64], elu+bias fused, then BN ---------------
    wmma_gemm_kernel<<<cdiv(N, 32 * MI), T, 0, stream>>>(H, 64, mlp1_w, 64,
                                                         AGG, 64, mlp1_b, 1, N, 64, 4, MI);
    zero_kernel<<<2, 192, 0, stream>>>(RED, 2 * 192);
    bn_stats_kernel<<<cdiv(N, 256), 64, 0, stream>>>(AGG, 64, N, sums, sumsq);
    bn_finalize_kernel<<<1, 64, 0, stream>>>(sums, sumsq, bnm1_g, bnm1_b, scale, shift, 64, N);
    bn_apply_kernel<<<cdiv(N64, T), T, 0, stream>>>(AGG, scale, shift, 64, N64);

    // --- mlp2: AGG[64] -> OUT[32], elu+bias fused, then BN (in place) --
    wmma_gemm_kernel<<<cdiv(N, 64 * MI), T, 0, stream>>>(AGG, 64, mlp2_w, 32,
                                                         OUT, 32, mlp2_b, 1, N, 64, 2, MI);
    zero_kernel<<<2, 192, 0, stream>>>(RED, 2 * 192);
    bn_stats_kernel<<<cdiv(N, 256), 32, 0, stream>>>(OUT, 32, N, sums, sumsq);
    bn_finalize_kernel<<<1, 32, 0, stream>>>(sums, sumsq, bnm2_g, bnm2_b, scale, shift, 32, N);
    bn_apply_kernel<<<cdiv((long)N * 32, T), T, 0, stream>>>(OUT, scale, shift, 32, (long)N * 32);
}